// Ranker_47820165874362
// MI455X (gfx1250) — compile-verified
//
#include <hip/hip_runtime.h>
#include <hip/hip_bf16.h>

typedef float v2f_t __attribute__((ext_vector_type(2)));
typedef float v8f_t __attribute__((ext_vector_type(8)));

#define BATCH 64
#define DDIM  3840
#define NB    32768
#define NNZ_E 1048576

// ---------------------------------------------------------------------------
// Kernel 1: zero the aggregation scratch (aggT is [NB, 64] f32 = 8 MB)
// ---------------------------------------------------------------------------
__global__ void zero_kernel(float* __restrict__ p, int n) {
    int i = blockIdx.x * blockDim.x + threadIdx.x;
    if (i < n) p[i] = 0.0f;
}

// ---------------------------------------------------------------------------
// Kernel 2: fused dual GEMM with V_WMMA_F32_16X16X4_F32.
//   out[b,n] = A[b,:] @ W1[:,n] + b1[n]      -> written to d_out
//   zT[n,b]  = A[b,:] @ Wa[:,n] + ba[n]      -> written to workspace (transposed)
//
// Block = 256 threads = 8 waves. Wave w: m_tile = w&3 (4 tiles cover B=64),
// n_tile = blockIdx.x*2 + (w>>2). Grid = NB/16/2 = 1024 blocks.
// Each wave runs K/4 = 960 iterations, 2 WMMAs each (shared A fragment).
// ---------------------------------------------------------------------------
__global__ void __launch_bounds__(256, 4)
gemm_wmma_kernel(const float* __restrict__ A,   // [64, 3840]
                 const float* __restrict__ W1,  // [3840, 32768]
                 const float* __restrict__ b1,  // [32768]
                 const float* __restrict__ Wa,  // [3840, 32768]
                 const float* __restrict__ ba,  // [32768]
                 float* __restrict__ out,       // [64, 32768]
                 float* __restrict__ zT)        // [32768, 64]
{
    const int lane  = threadIdx.x & 31;
    const int wid   = threadIdx.x >> 5;
    const int mTile = wid & 3;                     // 0..3  (rows of batch)
    const int nTile = blockIdx.x * 2 + (wid >> 2); // 0..2047

    const int l15  = lane & 15;
    const int lhi  = lane >> 4;                    // 0 or 1 (lane half)

    // A fragment addressing: lane half selects K pair {0,1} vs {2,3}
    const int mRow = mTile * 16 + l15;
    const int kOff = lhi * 2;
    const float* Arow = A + (size_t)mRow * DDIM + kOff;

    // B fragment addressing: VGPR0 -> row k+lhi, VGPR1 -> row k+2+lhi, col n
    const int col = nTile * 16 + l15;
    const float* W1p = W1 + (size_t)lhi * NB + col;
    const float* Wap = Wa + (size_t)lhi * NB + col;

    const float bias1 = b1[col];
    const float biasA = ba[col];
    v8f_t acc1 = {bias1, bias1, bias1, bias1, bias1, bias1, bias1, bias1};
    v8f_t acc2 = {biasA, biasA, biasA, biasA, biasA, biasA, biasA, biasA};

    #pragma unroll 4
    for (int k = 0; k < DDIM; k += 4) {
        // A 16x4 fragment: 2 consecutive f32 per lane (8B aligned)
        v2f_t afrag = *(const v2f_t*)(Arow + k);

        // W1 4x16 fragment
        v2f_t bf1;
        bf1.x = W1p[(size_t)k * NB];
        bf1.y = W1p[(size_t)(k + 2) * NB];

        // Wa 4x16 fragment
        v2f_t bfa;
        bfa.x = Wap[(size_t)k * NB];
        bfa.y = Wap[(size_t)(k + 2) * NB];

        // D = A*B + C  (8 args: neg_a, A, neg_b, B, c_mod, C, reuse_a, reuse_b)
        acc1 = __builtin_amdgcn_wmma_f32_16x16x4_f32(
            false, afrag, false, bf1, (short)0, acc1, false, false);
        acc2 = __builtin_amdgcn_wmma_f32_16x16x4_f32(
            false, afrag, false, bfa, (short)0, acc2, false, false);
    }

    // C/D layout: VGPR i -> M = i (lanes 0-15) or M = 8+i (lanes 16-31), N = lane&15
    #pragma unroll
    for (int i = 0; i < 8; ++i) {
        int row = mTile * 16 + i + (lhi << 3);
        out[(size_t)row * NB + col] = acc1[i];
        zT[(size_t)col * BATCH + row] = acc2[i];
    }
}

// ---------------------------------------------------------------------------
// Kernel 3: COO SpMM scatter. 64 threads per edge e:
//   aggT[rows[e], b] += vals[e] * zT[cols[e], b]    (b = 0..63, coalesced)
// ---------------------------------------------------------------------------
__global__ void __launch_bounds__(256)
spmm_kernel(const float* __restrict__ vals,
            const int*   __restrict__ rows,
            const int*   __restrict__ cols,
            const float* __restrict__ zT,    // [NB, 64]
            float*       __restrict__ aggT,  // [NB, 64]
            int nnz)
{
    const int e = blockIdx.x * (blockDim.x >> 6) + (threadIdx.x >> 6);
    const int b = threadIdx.x & 63;
    if (e < nnz) {
        const int   r = rows[e];
        const int   c = cols[e];
        const float v = vals[e];
        float contrib = v * zT[(size_t)c * BATCH + b];
        unsafeAtomicAdd(&aggT[(size_t)r * BATCH + b], contrib);
    }
}

// ---------------------------------------------------------------------------
// Kernel 4: out[b,n] += ra_layers * aggT[n,b]
// ---------------------------------------------------------------------------
__global__ void __launch_bounds__(256)
final_add_kernel(float* __restrict__ out,
                 const float* __restrict__ aggT,
                 const int* __restrict__ ra_layers)
{
    const int idx = blockIdx.x * blockDim.x + threadIdx.x;  // over 64*32768
    const float f = (float)(*ra_layers);
    const int b = idx >> 15;        // / NB
    const int n = idx & (NB - 1);   // % NB
    out[idx] += f * aggT[(size_t)n * BATCH + b];
}

// ---------------------------------------------------------------------------
extern "C" void kernel_launch(void* const* d_in, const int* in_sizes, int n_in,
                              void* d_out, int out_size, void* d_ws, size_t ws_size,
                              hipStream_t stream) {
    const float* A    = (const float*)d_in[0];  // input_feature [64,3840]
    const float* W1   = (const float*)d_in[1];  // [3840,32768]
    const float* b1   = (const float*)d_in[2];  // [32768]
    const float* Wa   = (const float*)d_in[3];  // [3840,32768]
    const float* ba   = (const float*)d_in[4];  // [32768]
    const float* vals = (const float*)d_in[5];  // [NNZ]
    const int*   rows = (const int*)d_in[6];    // [NNZ]
    const int*   cols = (const int*)d_in[7];    // [NNZ]
    const int*   ra   = (const int*)d_in[8];    // scalar ra_layers

    float* out  = (float*)d_out;
    float* zT   = (float*)d_ws;                              // 8 MB: [NB,64]
    float* aggT = (float*)d_ws + (size_t)NB * BATCH;         // 8 MB: [NB,64]

    // 1) zero agg scratch (harness poisons ws; atomics need zeros each call)
    {
        int n = NB * BATCH;
        zero_kernel<<<(n + 255) / 256, 256, 0, stream>>>(aggT, n);
    }
    // 2) fused dual GEMM: 1024 blocks x 8 waves, one 16x16 tile per wave
    gemm_wmma_kernel<<<NB / 32, 256, 0, stream>>>(A, W1, b1, Wa, ba, out, zT);
    // 3) SpMM scatter: 4 edges per 256-thread block
    spmm_kernel<<<NNZ_E / 4, 256, 0, stream>>>(vals, rows, cols, zT, aggT, NNZ_E);
    // 4) out += ra_layers * agg
    final_add_kernel<<<(BATCH * NB) / 256, 256, 0, stream>>>(out, aggT, ra);
}